// ScaffoldGNNEncoder_12051678233105
// MI455X (gfx1250) — compile-verified
//
#include <hip/hip_runtime.h>
#include <hip/hip_bf16.h>

#define H_DIM 128
#define G_NUM 256

typedef __attribute__((ext_vector_type(2))) float v2f;
typedef __attribute__((ext_vector_type(8))) float v8f;

// ---------------- small utility kernels ----------------

__global__ void set_value_kernel(float* __restrict__ p, float v, int n) {
    int i = blockIdx.x * blockDim.x + threadIdx.x;
    if (i < n) p[i] = v;
}

__global__ void deg_accum_kernel(const int* __restrict__ col, float* __restrict__ deg, int E) {
    int e = blockIdx.x * blockDim.x + threadIdx.x;
    if (e < E) atomicAdd(&deg[col[e]], 1.0f);
}

__global__ void dinv_kernel(const float* __restrict__ deg, float* __restrict__ dinv, int n) {
    int i = blockIdx.x * blockDim.x + threadIdx.x;
    if (i < n) {
        float d = deg[i];
        dinv[i] = d > 0.0f ? rsqrtf(d) : 0.0f;
    }
}

// ---------------- WMMA GEMM: Out[M,NCOLS] = X[M,128] @ W[128,NCOLS] (+bias) ----------------
// One wave computes a 16x16 tile. Block = NCOLS/16 waves covering one 16-row strip.

template <int NCOLS>
__global__ void gemm_k128_wmma(const float* __restrict__ X, const float* __restrict__ W,
                               const float* __restrict__ bias, float* __restrict__ Out, int M) {
    __shared__ float xs[16 * 132];  // 16 rows x 128 cols, padded stride 132 (bank spread)

    const int r0 = blockIdx.x * 16;

    // Cooperatively stage the 16x128 X strip into LDS.
    for (int idx = threadIdx.x; idx < 16 * H_DIM; idx += blockDim.x) {
        int r = idx >> 7;
        int c = idx & 127;
        float v = 0.0f;
        int gr = r0 + r;
        if (gr < M) v = X[(long)gr * H_DIM + c];
        xs[r * 132 + c] = v;
    }
    __syncthreads();

    const int wave = threadIdx.x >> 5;
    const int lane = threadIdx.x & 31;
    const int half = lane >> 4;   // 0: lanes 0-15, 1: lanes 16-31
    const int lr   = lane & 15;
    const int col0 = wave * 16;

    v8f acc = {};
#pragma unroll
    for (int k0 = 0; k0 < H_DIM; k0 += 4) {
        const int ka = k0 + 2 * half;
        // A 16x4 f32 fragment: lanes 0-15 carry K=k0,k0+1; lanes 16-31 carry K=k0+2,k0+3.
        v2f a;
        a[0] = xs[lr * 132 + ka];
        a[1] = xs[lr * 132 + ka + 1];
        // B 4x16 f32 fragment (mirrored layout), W row-major [128][NCOLS].
        v2f b;
        b[0] = W[(long)ka * NCOLS + col0 + lr];
        b[1] = W[(long)(ka + 1) * NCOLS + col0 + lr];
        acc = __builtin_amdgcn_wmma_f32_16x16x4_f32(false, a, false, b, (short)0, acc,
                                                    false, false);
    }

    const float bv = bias ? bias[col0 + lr] : 0.0f;
    // D layout: VGPR v -> row r0 + v + 8*half, column col0 + lr.
#pragma unroll
    for (int v = 0; v < 8; v++) {
        int r = r0 + v + 8 * half;
        if (r < M) Out[(long)r * NCOLS + col0 + lr] = acc[v] + bv;
    }
}

// ---------------- edge scatter: agg[col] += dinv[row]*dinv[col] * hT[row] ----------------
// Block = 256 threads = 2 edges x 128 features.

__global__ void edge_scatter_kernel(const float* __restrict__ hT, const float* __restrict__ dinv,
                                    const int* __restrict__ row, const int* __restrict__ col,
                                    float* __restrict__ agg, int E) {
    int e = blockIdx.x * 2 + (threadIdx.x >> 7);
    int c = threadIdx.x & 127;
    if (e < E) {
        int r = row[e];
        int t = col[e];
        float norm = dinv[r] * dinv[t];
        atomicAdd(&agg[(long)t * H_DIM + c], norm * hT[(long)r * H_DIM + c]);
    }
}

// ---------------- finalize: out = relu(agg + dinv^2*hT + b) (self-loop folded in) ----------

__global__ void finalize_kernel(const float* __restrict__ agg, const float* __restrict__ hT,
                                const float* __restrict__ dinv, const float* __restrict__ b,
                                float* __restrict__ out, int n) {
    long i = (long)blockIdx.x * blockDim.x + threadIdx.x;
    if (i < (long)n * H_DIM) {
        int node = (int)(i >> 7);
        int c = (int)(i & 127);
        float di = dinv[node];
        float v = agg[i] + di * di * hT[i] + b[c];
        out[i] = v > 0.0f ? v : 0.0f;
    }
}

// ---------------- pooling ----------------

__global__ void pool_accum_kernel(const float* __restrict__ h, const int* __restrict__ batch,
                                  float* __restrict__ sums, float* __restrict__ cnt, int n) {
    int i = blockIdx.x;
    int c = threadIdx.x;  // 128 threads
    if (i < n) {
        int g = batch[i];
        atomicAdd(&sums[(long)g * H_DIM + c], h[(long)i * H_DIM + c]);
        if (c == 0) atomicAdd(&cnt[g], 1.0f);
    }
}

__global__ void pool_div_kernel(float* __restrict__ sums, const float* __restrict__ cnt, int g_total) {
    int i = blockIdx.x * blockDim.x + threadIdx.x;
    if (i < g_total * H_DIM) {
        int g = i >> 7;
        float c = cnt[g];
        c = c > 1.0f ? c : 1.0f;
        sums[i] /= c;
    }
}

// ---------------- launch ----------------

extern "C" void kernel_launch(void* const* d_in, const int* in_sizes, int n_in,
                              void* d_out, int out_size, void* d_ws, size_t ws_size,
                              hipStream_t stream) {
    (void)n_in; (void)out_size; (void)ws_size;

    const float* x   = (const float*)d_in[0];
    const float* W1  = (const float*)d_in[1];
    const float* b1  = (const float*)d_in[2];
    const float* W2  = (const float*)d_in[3];
    const float* b2  = (const float*)d_in[4];
    const float* W3  = (const float*)d_in[5];
    const float* b3  = (const float*)d_in[6];
    const float* Wfc = (const float*)d_in[7];
    const float* bfc = (const float*)d_in[8];
    const int* edge  = (const int*)d_in[9];
    const int* batch = (const int*)d_in[10];

    const int E = in_sizes[9] / 2;
    const int N = in_sizes[10];
    const int* row = edge;       // edge_index[0, :] (sources)
    const int* col = edge + E;   // edge_index[1, :] (targets)

    float* ws = (float*)d_ws;
    auto alignup = [](size_t v) { return (v + 63) & ~(size_t)63; };
    size_t o = 0;
    float* deg  = ws + o; o = alignup(o + (size_t)N);
    float* dinv = ws + o; o = alignup(o + (size_t)N);
    float* bufT = ws + o; o = alignup(o + (size_t)N * H_DIM);
    float* bufA = ws + o; o = alignup(o + (size_t)N * H_DIM);
    float* agg  = ws + o; o = alignup(o + (size_t)N * H_DIM);
    float* sums = ws + o; o = alignup(o + (size_t)G_NUM * H_DIM);
    float* cnt  = ws + o; o = alignup(o + (size_t)G_NUM);

    // --- degree / symmetric norm (self-loop contributes deg=1) ---
    set_value_kernel<<<(N + 255) / 256, 256, 0, stream>>>(deg, 1.0f, N);
    deg_accum_kernel<<<(E + 255) / 256, 256, 0, stream>>>(col, deg, E);
    dinv_kernel<<<(N + 255) / 256, 256, 0, stream>>>(deg, dinv, N);

    // --- 3 GCN layers ---
    const float* hin = x;
    const float* Ws[3] = {W1, W2, W3};
    const float* bs[3] = {b1, b2, b3};
    for (int l = 0; l < 3; l++) {
        gemm_k128_wmma<H_DIM><<<(N + 15) / 16, 256, 0, stream>>>(hin, Ws[l], nullptr, bufT, N);
        hipMemsetAsync(agg, 0, (size_t)N * H_DIM * sizeof(float), stream);
        edge_scatter_kernel<<<(E + 1) / 2, 256, 0, stream>>>(bufT, dinv, row, col, agg, E);
        finalize_kernel<<<(int)(((long)N * H_DIM + 255) / 256), 256, 0, stream>>>(
            agg, bufT, dinv, bs[l], bufA, N);
        hin = bufA;
    }

    // --- global mean pool ---
    hipMemsetAsync(sums, 0, (size_t)G_NUM * H_DIM * sizeof(float), stream);
    hipMemsetAsync(cnt, 0, (size_t)G_NUM * sizeof(float), stream);
    pool_accum_kernel<<<N, 128, 0, stream>>>(bufA, batch, sums, cnt, N);
    pool_div_kernel<<<(G_NUM * H_DIM + 255) / 256, 256, 0, stream>>>(sums, cnt, G_NUM);

    // --- FC head: out[G, 256] = pooled @ Wfc + bfc ---
    gemm_k128_wmma<256><<<(G_NUM + 15) / 16, 512, 0, stream>>>(sums, Wfc, bfc, (float*)d_out, G_NUM);
}